// DifferentiableSoftmaxTree_71966472012140
// MI455X (gfx1250) — compile-verified
//
#include <hip/hip_runtime.h>
#include <math.h>
#include <stdint.h>

typedef float v2f __attribute__((ext_vector_type(2)));
typedef float v8f __attribute__((ext_vector_type(8)));
typedef unsigned int u32x4 __attribute__((ext_vector_type(4)));
typedef int i32x4 __attribute__((ext_vector_type(4)));
typedef int i32x8 __attribute__((ext_vector_type(8)));

#define FEATURE_DIM 512
#define TREE_DEPTH  16
#define NUM_INTERNAL 49999

// LDS layout (float indices). Dynamic shared memory starts at LDS offset 0.
#define W_F     0        // 16 nodes * 1024 floats ([d][c] interleaved) = 64KB
#define F_F     16384    // 512 floats feature vector
#define NODE_I  16896    // 16 ints (as int index into smem)
#define DIR_I   16912    // 16 ints
#define PART_F  16928    // 4 partial sums
#define LDS_FLOATS 16932

__global__ __launch_bounds__(64)
void hsm_tree_kernel(const float* __restrict__ features,
                     const int*   __restrict__ targets,
                     const float* __restrict__ node_weights,
                     const int*   __restrict__ path_nodes,
                     const int*   __restrict__ path_dirs,
                     float*       __restrict__ out)
{
    extern __shared__ float smem[];
    int* smi = (int*)smem;

    const int b    = blockIdx.x;
    const int tid  = threadIdx.x;
    const int wave = tid >> 5;      // 0..1, one 16-row WMMA tile each
    const int lane = tid & 31;
    const int t    = targets[b];

    // Path metadata -> LDS
    if (tid < TREE_DEPTH) {
        smi[NODE_I + tid] = path_nodes[t * TREE_DEPTH + tid];
        smi[DIR_I  + tid] = path_dirs [t * TREE_DEPTH + tid];
    }

    // ---- TDM gather: 16 weight rows (4KB each) -> LDS in ONE tensor op ----
    // D# gather mode, 16-bit row indices (node ids < 50000 fit; strictly
    // increasing along the path, padding clamped to 0).
    if (wave == 0) {
        int n[16];
        #pragma unroll
        for (int k = 0; k < TREE_DEPTH; ++k) {
            int v = path_nodes[t * TREE_DEPTH + k];
            n[k] = (v < 0) ? 0 : v;
        }
        const uint64_t ga = (uint64_t)node_weights;
        u32x4 g0;
        g0.x = 0x80000001u;                              // count=1, gather_mode=1, 16b idx
        g0.y = 0u;                                       // lds_addr = 0 (W_F region)
        g0.z = (unsigned)(ga & 0xffffffffu);             // global_addr[31:0]
        g0.w = (unsigned)((ga >> 32) & 0x01ffffffu)      // global_addr[56:32]
             | 0x80000000u;                              // type = 2 ("image")
        i32x8 g1;
        g1[0] = 0x00020000;                              // mask=0, data_size=4B
        g1[1] = (int)(1024u << 16);                      // tensor_dim0 = 1024
        g1[2] = (int)(((unsigned)NUM_INTERNAL) << 16);   // tensor_dim1 = 49999
        g1[3] = (int)(1024u << 16);                      // tile_dim0 = 1024
        g1[4] = TREE_DEPTH;                              // tile_dim1 = 16 valid indices
        g1[5] = 1024;                                    // tensor_dim0_stride = 1024
        g1[6] = 0;
        g1[7] = 0;
        i32x4 g2, g3;                                    // 16-bit packed row indices
        #pragma unroll
        for (int j = 0; j < 4; ++j) {
            g2[j] = (n[2*j] & 0xffff) | (n[2*j + 1] << 16);
            g3[j] = (n[8 + 2*j] & 0xffff) | (n[8 + 2*j + 1] << 16);
        }
#if defined(__clang_major__) && (__clang_major__ >= 23)
        i32x8 gz = {};
        __builtin_amdgcn_tensor_load_to_lds(g0, g1, g2, g3, gz, 0);
#else
        __builtin_amdgcn_tensor_load_to_lds(g0, g1, g2, g3, 0);
#endif
    }

    // ---- Async stage: feature vector (2KB) global -> LDS ----
    const unsigned voff = (unsigned)lane * 16u;
    for (int i = 0; i < 2; ++i) {
        const int cc = wave * 2 + i;          // chunk 0..3
        const uint64_t g = (uint64_t)features
                         + (uint64_t)b * (FEATURE_DIM * 4u) + (uint64_t)cc * 512u;
        const unsigned lds_off = (unsigned)(F_F * 4) + (unsigned)cc * 512u + voff;
        asm volatile("global_load_async_to_lds_b128 %0, %1, %2"
                     :: "v"(lds_off), "v"(voff), "s"(g) : "memory");
    }
    asm volatile("s_wait_asynccnt 0" ::: "memory");
    if (wave == 0) __builtin_amdgcn_s_wait_tensorcnt(0);
    __syncthreads();

    // ---- WMMA f32 16x16x4 matvec: rows = (node,dir) pairs, K chains over 512 ----
    // Tile for this wave: global rows 16*wave + m, m = lane&15.
    // Row r = 2k+c maps to node slot k, column c of node_weights[k][d][c].
    const int m    = lane & 15;
    const int hi   = lane >> 4;                    // A: K=2hi..2hi+1 in vgpr0/1
    const int krow = ((wave << 4) + m) >> 1;       // node slot for this row
    const int cdir = m & 1;
    const int abase = krow * 1024 + 4 * hi + cdir; // + 2*d0 -> W[row][d0+2hi]

    v8f acc0 = {}, acc1 = {}, acc2 = {}, acc3 = {};
    for (int d0 = 0; d0 < FEATURE_DIM; d0 += 16) {
        #pragma unroll
        for (int j = 0; j < 4; ++j) {
            const int d = d0 + 4 * j;
            v2f a;
            a.x = smem[W_F + abase + 2 * d];        // W[row][d + 2*hi]
            a.y = smem[W_F + abase + 2 * d + 2];    // W[row][d + 2*hi + 1]
            // B 4x16: only column 0 used -> lanes 0 (K0,K1) and 16 (K2,K3)
            const float fx = smem[F_F + d + 2 * hi];
            const float fy = smem[F_F + d + 2 * hi + 1];
            v2f bv;
            bv.x = (m == 0) ? fx : 0.0f;
            bv.y = (m == 0) ? fy : 0.0f;
            v8f c = (j == 0) ? acc0 : (j == 1) ? acc1 : (j == 2) ? acc2 : acc3;
            c = __builtin_amdgcn_wmma_f32_16x16x4_f32(
                    false, a, false, bv, (short)0, c, false, false);
            if (j == 0) acc0 = c; else if (j == 1) acc1 = c;
            else if (j == 2) acc2 = c; else acc3 = c;
        }
    }
    v8f r;
    #pragma unroll
    for (int e = 0; e < 8; ++e) r[e] = acc0[e] + acc1[e] + acc2[e] + acc3[e];

    // ---- log-softmax + select: column 0 lives in lanes 0 (rows 0-7) / 16 (rows 8-15)
    if (m == 0) {
        float psum = 0.0f;
        #pragma unroll
        for (int j = 0; j < 4; ++j) {
            const int k = (wave << 3) + (hi << 2) + j;   // node slot 0..15
            const int node = smi[NODE_I + k];
            if (node >= 0) {
                const float l0 = r[2 * j];
                const float l1 = r[2 * j + 1];
                const int  dir = smi[DIR_I + k];
                const float mx  = fmaxf(l0, l1);
                const float lse = mx + logf(expf(l0 - mx) + expf(l1 - mx));
                psum += (dir ? l1 : l0) - lse;
            }
        }
        smem[PART_F + wave * 2 + hi] = psum;
    }
    __syncthreads();
    if (tid == 0) {
        out[b] = -(smem[PART_F + 0] + smem[PART_F + 1] +
                   smem[PART_F + 2] + smem[PART_F + 3]);
    }
}

extern "C" void kernel_launch(void* const* d_in, const int* in_sizes, int n_in,
                              void* d_out, int out_size, void* d_ws, size_t ws_size,
                              hipStream_t stream) {
    const float* features     = (const float*)d_in[0];
    const int*   targets      = (const int*)  d_in[1];
    const float* node_weights = (const float*)d_in[2];
    const int*   path_nodes   = (const int*)  d_in[3];
    const int*   path_dirs    = (const int*)  d_in[4];
    float* out = (float*)d_out;

    const int batch = in_sizes[0] / FEATURE_DIM;   // 4096
    const size_t shmem = (size_t)LDS_FLOATS * sizeof(float);  // ~67.7KB
    hsm_tree_kernel<<<dim3(batch), dim3(64), shmem, stream>>>(
        features, targets, node_weights, path_nodes, path_dirs, out);
}